// LSTM_36318243455555
// MI455X (gfx1250) — compile-verified
//
#include <hip/hip_runtime.h>

// LSTM forward on MI455X (gfx1250), fp32 end-to-end via V_WMMA_F32_16X16X4_F32.
// SEQ=512, B=64, D=H=1024.
//
// Phase 0 (parallel):  pack gate weights into WMMA-native layout:
//     Wp[k2][n][g] = (W_g[2*k2][n], W_g[2*k2+1][n])  -- float2 per gate
//   so one b128 load yields the B operand pairs for two gates.
// Phase 1 (parallel):  gx[g][m][j] = x @ Wx_g + bx_g + bh_g   (m = t*B+b)
// Phase 2 (persistent): per step t, each wave owns a 16x16 (m,j) tile and all
//   4 gates; h @ Wh_g via fp32 WMMA; gates fused; c kept in registers (C/D
//   tile layout) for the entire sequence; grid-wide atomic barrier per step.

#define SEQL 512
#define BATCH 64
#define DIM 1024
#define HID 1024

typedef __attribute__((ext_vector_type(2))) float v2f;
typedef __attribute__((ext_vector_type(4))) float v4f;
typedef __attribute__((ext_vector_type(8))) float v8f;

__device__ __forceinline__ v8f zero8() {
  v8f z;
#pragma unroll
  for (int i = 0; i < 8; ++i) z[i] = 0.0f;
  return z;
}

// D = A(16x4, fp32) * B(4x16, fp32) + C(16x16, fp32); exact fp32 matrix op.
__device__ __forceinline__ v8f wmma4(v2f a, v2f b, v8f c) {
  return __builtin_amdgcn_wmma_f32_16x16x4_f32(
      /*neg_a=*/false, a, /*neg_b=*/false, b,
      /*c_mod=*/(short)0, c, /*reuse_a=*/false, /*reuse_b=*/false);
}

__device__ __forceinline__ float fsigmoid(float x) {
  return 1.0f / (1.0f + __expf(-x));
}
__device__ __forceinline__ float ftanh(float x) {
  // tanh(x) = 1 - 2/(e^{2x}+1); saturates correctly at +-1 via inf/0.
  return 1.0f - 2.0f / (__expf(2.0f * x) + 1.0f);
}

// ---------------- Phase 0: weight packing -----------------------------------
// Wp[((k2*HID + n)*4 + g)*2 + c] = W_g[(2*k2 + c)*HID + n]
// One thread per (k2, n): 8 strided-coalesced b32 reads, 2 contiguous b128
// writes. Run once per launch for Wx and Wh (32 MB total traffic).
__global__ __launch_bounds__(256) void lstm_pack(const float* __restrict__ W0,
                                                 const float* __restrict__ W1,
                                                 const float* __restrict__ W2,
                                                 const float* __restrict__ W3,
                                                 float* __restrict__ Wp) {
  const int idx = blockIdx.x * 256 + threadIdx.x;  // 0 .. (HID/2)*HID - 1
  const int n = idx & (HID - 1);
  const int k2 = idx >> 10;
  const size_t r0 = (size_t)(2 * k2) * HID + n;
  const size_t r1 = r0 + HID;
  v4f q01, q23;
  q01.x = W0[r0]; q01.y = W0[r1];
  q01.z = W1[r0]; q01.w = W1[r1];
  q23.x = W2[r0]; q23.y = W2[r1];
  q23.z = W3[r0]; q23.w = W3[r1];
  float* dst = Wp + (size_t)idx * 8;
  *(v4f*)dst = q01;
  *(v4f*)(dst + 4) = q23;
}

struct GxArgs {
  const float* x;      // [SEQ*B, D]
  const float* Wxp;    // packed [D/2][H][4][2]
  const float* bx[4];  // [H]
  const float* bh[4];  // [H]
  float* gx;           // [4][SEQ*B][H] gate-major, biases folded in
};

struct RecArgs {
  const float* gx;     // [4][SEQ*B][H]
  const float* Whp;    // packed [H/2][H][4][2]
  float* out;          // h_seq [SEQ][B][H] ++ h [B][H] ++ c [B][H]
  unsigned* sync;      // [0]=arrival counter, [1]=generation
};

// ---------------- Phase 1: x-projection GEMM --------------------------------
// Wave tile: 16 rows (m) x 16 cols (j) x 4 gates. grid = (64 ntiles, 512),
// 4 waves/block -> 2048 mtiles. Blocks along x share A rows (L2 reuse);
// the 16 MB of packed Wx stays L2-resident across the whole launch.
// Inner loop per K=4: 1x b64 (A) + 2x b128 (B, all four gates) + 4 WMMA.
__global__ __launch_bounds__(128) void lstm_gx_gemm(GxArgs p) {
  const int lane = threadIdx.x & 31;
  const int wv = threadIdx.x >> 5;
  const int hi = lane >> 4;   // K-half select (lanes 16-31 hold K=+2,+3)
  const int ln = lane & 15;
  const int mt = blockIdx.y * 4 + wv;  // 0..2047
  const int nt = blockIdx.x;           // 0..63
  const int m0 = mt * 16;
  const int n = nt * 16 + ln;

  v8f acc[4];
#pragma unroll
  for (int g = 0; g < 4; ++g) acc[g] = zero8();

  const float* arow = p.x + (size_t)(m0 + ln) * DIM + 2 * hi;
  // k2 = k/2 + hi  ->  float offset ((k2*HID)+n)*8 == k*HID*4 + (hi*HID+n)*8
  const float* bbase = p.Wxp + (size_t)(hi * HID + n) * 8;
#pragma unroll 2
  for (int k = 0; k < DIM; k += 4) {
    v2f a = *(const v2f*)(arow + k);  // A[m, kk], A[m, kk+1]
    const float* bp = bbase + (size_t)k * (HID * 4);
    v4f q01 = *(const v4f*)bp;        // gates f,i : (B[kk,n], B[kk+1,n])
    v4f q23 = *(const v4f*)(bp + 4);  // gates o,c
    acc[0] = wmma4(a, q01.xy, acc[0]);
    acc[1] = wmma4(a, q01.zw, acc[1]);
    acc[2] = wmma4(a, q23.xy, acc[2]);
    acc[3] = wmma4(a, q23.zw, acc[3]);
  }

  const size_t GXG = (size_t)SEQL * BATCH * HID;
#pragma unroll
  for (int g = 0; g < 4; ++g) {
    const float bsum = p.bx[g][n] + p.bh[g][n];
#pragma unroll
    for (int r = 0; r < 8; ++r) {
      const int m = m0 + r + 8 * hi;  // C/D layout: VGPR r holds rows r, r+8
      p.gx[(size_t)g * GXG + (size_t)m * HID + n] = acc[g][r] + bsum;
    }
  }
}

// ---------------- Phase 2: persistent recurrence ----------------------------
// 64 blocks x 128 threads = 256 waves; wave wid: mt = wid&3 (batch tile),
// jt = wid>>2 (hidden tile) -> the 4 waves of a block share jt => identical
// packed-Wh b128 reads hit L0/L2. c tile lives in registers for all 512 steps.
__global__ __launch_bounds__(128) void lstm_recurrent(RecArgs p) {
  const int lane = threadIdx.x & 31;
  const int wv = threadIdx.x >> 5;
  const int hi = lane >> 4;
  const int ln = lane & 15;
  const int wid = blockIdx.x * 4 + wv;  // 0..255
  const int mt = wid & 3;               // batch-row tile 0..3
  const int jt = wid >> 2;              // hidden-col tile 0..63
  const int m0 = mt * 16;
  const int n = jt * 16 + ln;
  const size_t GXG = (size_t)SEQL * BATCH * HID;
  const size_t HS = (size_t)BATCH * HID;
  const float* bbase = p.Whp + (size_t)(hi * HID + n) * 8;

  v8f creg = zero8();  // cell state tile, register-resident across all steps

  for (int t = 0; t < SEQL; ++t) {
    v8f acc[4];
#pragma unroll
    for (int g = 0; g < 4; ++g) acc[g] = zero8();

    if (t > 0) {  // h(-1) == 0 -> skip GEMM entirely on the first step
      const float* hrow =
          p.out + (size_t)(t - 1) * HS + (size_t)(m0 + ln) * HID + 2 * hi;
#pragma unroll 2
      for (int k = 0; k < HID; k += 4) {
        v2f a = *(const v2f*)(hrow + k);
        const float* bp = bbase + (size_t)k * (HID * 4);
        v4f q01 = *(const v4f*)bp;
        v4f q23 = *(const v4f*)(bp + 4);
        acc[0] = wmma4(a, q01.xy, acc[0]);
        acc[1] = wmma4(a, q01.zw, acc[1]);
        acc[2] = wmma4(a, q23.xy, acc[2]);
        acc[3] = wmma4(a, q23.zw, acc[3]);
      }
    }

    const size_t gxbase = ((size_t)t * BATCH) * HID;
    float* hout = p.out + (size_t)t * HS;
#pragma unroll
    for (int r = 0; r < 8; ++r) {
      const int m = m0 + r + 8 * hi;
      const size_t gxi = gxbase + (size_t)m * HID + n;
      const float pf = acc[0][r] + p.gx[gxi];
      const float pi = acc[1][r] + p.gx[GXG + gxi];
      const float po = acc[2][r] + p.gx[2 * GXG + gxi];
      const float pg = acc[3][r] + p.gx[3 * GXG + gxi];
      const float f = fsigmoid(pf);
      const float i = fsigmoid(pi);
      const float o = fsigmoid(po);
      const float g = ftanh(pg);
      const float c = f * creg[r] + i * g;
      creg[r] = c;
      const float h = o * ftanh(c);
      hout[(size_t)m * HID + n] = h;
      if (t == SEQL - 1) {  // final h, c slots of the tuple output
        p.out[(size_t)SEQL * HS + (size_t)m * HID + n] = h;
        p.out[(size_t)SEQL * HS + HS + (size_t)m * HID + n] = c;
      }
    }

    // ---- grid-wide barrier: h[t] visible before any wave reads it at t+1 ---
    __syncthreads();
    if (threadIdx.x == 0) {
      __threadfence();
      volatile unsigned* vgen = (volatile unsigned*)(p.sync + 1);
      const unsigned g0 = *vgen;
      if (atomicAdd(p.sync, 1u) == (unsigned)(gridDim.x - 1)) {
        p.sync[0] = 0u;
        __threadfence();
        *vgen = g0 + 1u;
      } else {
        while (*vgen == g0) { __builtin_amdgcn_s_sleep(1); }
      }
      __threadfence();
    }
    __syncthreads();
  }
}

__global__ void lstm_init(unsigned* sync) {
  if (threadIdx.x < 2) sync[threadIdx.x] = 0u;
}

extern "C" void kernel_launch(void* const* d_in, const int* in_sizes, int n_in,
                              void* d_out, int out_size, void* d_ws,
                              size_t ws_size, hipStream_t stream) {
  (void)in_sizes; (void)n_in; (void)out_size; (void)ws_size;
  // d_in: 0=x, then (Wxf,bxf,Wxi,bxi,Wxo,bxo,Wxc,bxc), (Whf,bhf,...,Whc,bhc)
  const float* Wx[4];
  const float* Wh[4];
  GxArgs gxa;
  gxa.x = (const float*)d_in[0];
  for (int g = 0; g < 4; ++g) {
    Wx[g] = (const float*)d_in[1 + 2 * g];
    gxa.bx[g] = (const float*)d_in[2 + 2 * g];
    gxa.bh[g] = (const float*)d_in[10 + 2 * g];
    Wh[g] = (const float*)d_in[9 + 2 * g];
  }

  // Workspace layout: gx (512 MB) | packed Wx (16 MB) | packed Wh (16 MB) | sync
  const size_t GX_ELEMS = (size_t)4 * SEQL * BATCH * HID;
  const size_t WP_ELEMS = (size_t)4 * HID * HID;
  float* gx = (float*)d_ws;
  float* Wxp = gx + GX_ELEMS;
  float* Whp = Wxp + WP_ELEMS;
  unsigned* sync = (unsigned*)(Whp + WP_ELEMS);
  gxa.Wxp = Wxp;
  gxa.gx = gx;

  RecArgs ra;
  ra.gx = gx;
  ra.Whp = Whp;
  ra.out = (float*)d_out;
  ra.sync = sync;

  lstm_init<<<1, 32, 0, stream>>>(sync);
  // Pack: one thread per (k2, n) pair -> (HID/2)*HID threads.
  const int packBlocks = (HID / 2) * HID / 256;
  lstm_pack<<<packBlocks, 256, 0, stream>>>(Wx[0], Wx[1], Wx[2], Wx[3], Wxp);
  lstm_pack<<<packBlocks, 256, 0, stream>>>(Wh[0], Wh[1], Wh[2], Wh[3], Whp);
  // 64 n-tiles (x) * 512 blocks (y) * 4 waves = 2048 m-tiles of 16 rows.
  lstm_gx_gemm<<<dim3(64, (SEQL * BATCH / 16) / 4), 128, 0, stream>>>(gxa);
  // Persistent: 64 blocks (trivially co-resident) * 4 waves = 256 tile-waves.
  lstm_recurrent<<<64, 128, 0, stream>>>(ra);
}